// TPSecond_18511309046143
// MI455X (gfx1250) — compile-verified
//
#include <hip/hip_runtime.h>

// ---------------------------------------------------------------------------
// TPSecond: batched equivariant tensor product, Z=65536 independent samples.
// Memory-bound (AI ~0.66 FLOP/B): per-z weights (14.2KB) streamed to LDS via
// gfx1250 async-LDS loads with staged s_wait_asynccnt overlap; small per-z
// mat-vecs executed on v_wmma_f32_16x16x4_f32 (broadcast-A trick, fp32).
// One wave (32 lanes) per z; 4 waves / 128-thread block; no cross-wave sync.
// ---------------------------------------------------------------------------

typedef float v2f __attribute__((ext_vector_type(2)));
typedef float v8f __attribute__((ext_vector_type(8)));
typedef int i32x4 __attribute__((__vector_size__(4 * sizeof(int))));
typedef int i32x2 __attribute__((__vector_size__(2 * sizeof(int))));

#define ZDIM_X1 78
#define ZDIM_X2 9
#define ZDIM_W  3564
#define ZDIM_OUT 108
#define WSTRIDE 3776            // per-wave LDS floats (weights + x + scratch)
#define WAVES_PER_BLOCK 4

// ---- gfx1250 feature detection (device pass only) -------------------------
#if defined(__HIP_DEVICE_COMPILE__) && defined(__gfx1250__)
  #if __has_builtin(__builtin_amdgcn_global_load_async_to_lds_b128) && \
      __has_builtin(__builtin_amdgcn_global_load_async_to_lds_b64)  && \
      __has_builtin(__builtin_amdgcn_global_load_async_to_lds_b32)  && \
      __has_builtin(__builtin_amdgcn_s_wait_asynccnt)
    #define USE_ASYNC 1
  #else
    #define USE_ASYNC 0
  #endif
  #if __has_builtin(__builtin_amdgcn_wmma_f32_16x16x4_f32)
    #define USE_WMMA 1
  #else
    #define USE_WMMA 0
  #endif
#else
  #define USE_ASYNC 0
  #define USE_WMMA 0
#endif

#if USE_ASYNC
#define GLOBAL_AS __attribute__((address_space(1)))
#define LDS_AS    __attribute__((address_space(3)))
// Builtin signatures (from clang diagnostics): arg0 = AS1 pointer to
// int-vector of payload size, arg1 = LDS pointer, arg2/arg3 = imm offset/cpol.
__device__ __forceinline__ void async_b128(const float* g, float* l) {
  __builtin_amdgcn_global_load_async_to_lds_b128(
      (GLOBAL_AS i32x4*)g, (LDS_AS i32x4*)l, 0, 0);
}
__device__ __forceinline__ void async_b64(const float* g, float* l) {
  __builtin_amdgcn_global_load_async_to_lds_b64(
      (GLOBAL_AS i32x2*)g, (LDS_AS i32x2*)l, 0, 0);
}
__device__ __forceinline__ void async_b32(const float* g, float* l) {
  __builtin_amdgcn_global_load_async_to_lds_b32(
      (GLOBAL_AS int*)g, (LDS_AS int*)l, 0, 0);
}
#define ASYNC_WAIT(n) do { __builtin_amdgcn_s_wait_asynccnt(n); \
                           asm volatile("" ::: "memory"); } while (0)
#else
#define ASYNC_WAIT(n) do { } while (0)
#endif

// ---------------------------------------------------------------------------
// wave_matvec: out[n] = scale * sum_{k<4*ksteps} xv[k] * wcol[k*ldw + n]
// Result valid on lanes 0..15 for column n = lane (lanes 16..31 duplicate).
// WMMA f32 16x16x4: A(16x4) rows all equal to the x chunk (lane-half selects
// K high bits, VGPR selects K low bit), B(4x16) = weight tile from LDS.
// ---------------------------------------------------------------------------
__device__ __forceinline__ float wave_matvec(const float* __restrict__ xv,
                                             const float* __restrict__ wcol,
                                             const int ldw, const int ksteps,
                                             const float scale, const int lane)
{
  const int hi = (lane >> 4) & 1;
  const int n  = lane & 15;
#if USE_WMMA
  v8f acc = {0.f, 0.f, 0.f, 0.f, 0.f, 0.f, 0.f, 0.f};
  #pragma unroll
  for (int ks = 0; ks < ksteps; ++ks) {
    const int ka = ks * 4 + 2 * hi;            // this lane-half's K pair
    const v2f xa = *(const v2f*)(xv + ka);     // ds_load_b64 (xv even-aligned)
    v2f a; a.x = scale * xa.x; a.y = scale * xa.y;
    v2f b; b.x = wcol[ka * ldw + n]; b.y = wcol[(ka + 1) * ldw + n];
    acc = __builtin_amdgcn_wmma_f32_16x16x4_f32(false, a, false, b,
                                                (short)0, acc, false, false);
  }
  return acc[0];
#else
  float s = 0.f;
  #pragma unroll
  for (int k = 0; k < ksteps * 4; ++k) s += xv[k] * wcol[k * ldw + n];
  return scale * s;
#endif
}

__global__ __launch_bounds__(WAVES_PER_BLOCK * 32)
void tp2_kernel(const float* __restrict__ gX1all,
                const float* __restrict__ gX2all,
                const float* __restrict__ gWall,
                float* __restrict__ gOut, const int Z)
{
  __shared__ float lds[WAVES_PER_BLOCK * WSTRIDE];   // 60416 B
  const int lane = threadIdx.x & 31;
  const int wv   = threadIdx.x >> 5;
  const int z    = blockIdx.x * WAVES_PER_BLOCK + wv;
  if (z >= Z) return;                                 // wave-uniform

  // per-wave LDS layout (float offsets; all xv bases even -> b64-aligned)
  float* const L   = lds + wv * WSTRIDE;
  float* const LW  = L;            // 0    .. 3563  weights
  float* const LX1 = L + 3564;     // 3564 .. 3641  x1 (48 + 10*3)
  float* const LX2 = L + 3642;     // 3642 .. 3650  x2 (1+3+5)
  float* const Lt  = L + 3652;     // 12: dot(v_u, r)           (pad 0)
  float* const LvL = L + 3664;     // 3x12: v components        (pad 0)
  float* const Ly1 = L + 3700;     // 3x12: cross(v_u, r)       (pad 0)
  float* const Ly2 = L + 3736;     // 3x12: (S v_u) permuted    (pad 0)

  const float* const gX1 = gX1all + (size_t)z * ZDIM_X1;
  const float* const gX2 = gX2all + (size_t)z * ZDIM_X2;
  const float* const gW  = gWall  + (size_t)z * ZDIM_W;

  // ---- stage x then weights (issue order defines the wait schedule) -------
#if USE_ASYNC
  {
    int o = lane * 2;
    async_b64(gX1 + o, LX1 + o);                       // x1[0..63]
    o = 64 + lane * 2;
    if (o < ZDIM_X1) async_b64(gX1 + o, LX1 + o);      // x1[64..77]
    if (lane < ZDIM_X2) async_b32(gX2 + lane, LX2 + lane);
    #pragma unroll
    for (int c = 0; c < 28; ++c) {                     // 28 x 512B chunks
      const int off = c * 128 + lane * 4;
      if (off < ZDIM_W) async_b128(gW + off, LW + off);
    }
  }
#else
  for (int i = lane; i < ZDIM_X1; i += 32) LX1[i] = gX1[i];
  if (lane < ZDIM_X2) LX2[lane] = gX2[lane];
  for (int i = lane * 4; i < ZDIM_W; i += 128) {       // 16B-aligned (z*14256)
    const float4 v4 = *(const float4*)(gW + i);
    *(float4*)(LW + i) = v4;
  }
#endif

  ASYNC_WAIT(28);                       // x1/x2 resident (3 of 31 complete)

  // ---- derived K=10 vectors (zero-padded to 12), same-wave DS ordering ----
  if (lane < 12) {
    float tt = 0.f, vx = 0.f, vy = 0.f, vz = 0.f;
    float y1a = 0.f, y1b = 0.f, y1c = 0.f, y2a = 0.f, y2b = 0.f, y2c = 0.f;
    if (lane < 10) {
      vx = LX1[48 + lane * 3 + 0];
      vy = LX1[48 + lane * 3 + 1];
      vz = LX1[48 + lane * 3 + 2];
      const float rx = LX2[1], ry = LX2[2], rz = LX2[3];
      tt  = vx * rx + vy * ry + vz * rz;               // 1x1->0 coupling
      y1a = vy * rz - vz * ry;                         // 1x1->1: cross(v, r)
      y1b = vz * rx - vx * rz;
      y1c = vx * ry - vy * rx;
      // 1x2->1: rebuild cartesian S from 5 B-basis comps, apply to v (y,z,x)
      const float e0 = LX2[4], e1 = LX2[5], e2 = LX2[6], e3 = LX2[7], e4 = LX2[8];
      const float is2 = 0.70710678118654752f;          // 1/sqrt(2)
      const float is6 = 0.40824829046386302f;          // 1/sqrt(6)
      const float Sxx = -e2 * is6 + e4 * is2;
      const float Syy = -e2 * is6 - e4 * is2;
      const float Szz =  2.f * e2 * is6;
      const float Sxy = e0 * is2, Syz = e1 * is2, Sxz = e3 * is2;
      const float Vx = vz, Vy = vx, Vz = vy;           // stored(y,z,x)->cart
      const float Ox = Sxx * Vx + Sxy * Vy + Sxz * Vz;
      const float Oy = Sxy * Vx + Syy * Vy + Syz * Vz;
      const float Oz = Sxz * Vx + Syz * Vy + Szz * Vz;
      y2a = Oy; y2b = Oz; y2c = Ox;                    // cart->stored(y,z,x)
    }
    Lt[lane] = tt;
    LvL[0 * 12 + lane] = vx;  LvL[1 * 12 + lane] = vy;  LvL[2 * 12 + lane] = vz;
    Ly1[0 * 12 + lane] = y1a; Ly1[1 * 12 + lane] = y1b; Ly1[2 * 12 + lane] = y1c;
    Ly2[0 * 12 + lane] = y2a; Ly2[1 * 12 + lane] = y2b; Ly2[2 * 12 + lane] = y2c;
  }

  const float c0     = 0.13130643f;   // sqrt(1/58)
  const float c1     = 0.21004201f;   // sqrt(3/68)
  const float c2     = 0.54772256f;   // sqrt(3/10)
  const float inv_s3 = 0.57735027f;
  const float inv_s5 = 0.44721360f;
  const float inv_s6 = 0.40824829f;

  const float x20 = LX2[0];
  const float r1x = LX2[1], r1y = LX2[2], r1z = LX2[3];

  float o0[3], o1[3], o2[3];

  ASYNC_WAIT(10);                      // w00 (first 18 weight chunks) resident
  #pragma unroll
  for (int t = 0; t < 3; ++t)          // r0e  = c0*x2_0e * (x1_0e . W00)
    o0[t] = wave_matvec(LX1, LW + t * 16, 48, 12, c0 * x20, lane);

  ASYNC_WAIT(6);                       // w01 resident
  {
    const float sw = wave_matvec(LX1, LW + 2304, 10, 12, c1 * inv_s3, lane);
    o1[0] = sw * r1x; o1[1] = sw * r1y; o1[2] = sw * r1z;
  }

  ASYNC_WAIT(5);                       // w10 resident
  #pragma unroll
  for (int i = 0; i < 3; ++i)          // r1o += (c1/s3)*x2_0e * (v_i . W10)
    o1[i] += wave_matvec(LvL + 12 * i, LW + 2784, 10, 3, c1 * inv_s3 * x20, lane);

  ASYNC_WAIT(1);                       // w110 resident
  #pragma unroll
  for (int t = 0; t < 3; ++t)          // r0e += (c0/s3) * (t . W110)
    o0[t] += wave_matvec(Lt, LW + 2884 + t * 16, 48, 3, c0 * inv_s3, lane);

  ASYNC_WAIT(0);                       // w112, w12 resident
  #pragma unroll
  for (int k = 0; k < 3; ++k)          // r1e  = (c2/s6) * (cross_k . W112)
    o2[k] = wave_matvec(Ly1 + 12 * k, LW + 3364, 10, 3, c2 * inv_s6, lane);
  #pragma unroll
  for (int k = 0; k < 3; ++k)          // r1o += (c1/s5) * ((Sv)_k . W12)
    o1[k] += wave_matvec(Ly2 + 12 * k, LW + 3464, 10, 3, c1 * inv_s5, lane);

  // ---- store: out = [r0e(48) | r1o(10x3) | r1e(10x3)] ---------------------
  float* const gO = gOut + (size_t)z * ZDIM_OUT;
  if (lane < 16) {
    gO[lane]      = o0[0];
    gO[16 + lane] = o0[1];
    gO[32 + lane] = o0[2];
    if (lane < 10) {
      gO[48 + lane * 3 + 0] = o1[0];
      gO[48 + lane * 3 + 1] = o1[1];
      gO[48 + lane * 3 + 2] = o1[2];
      gO[78 + lane * 3 + 0] = o2[0];
      gO[78 + lane * 3 + 1] = o2[1];
      gO[78 + lane * 3 + 2] = o2[2];
    }
  }
}

extern "C" void kernel_launch(void* const* d_in, const int* in_sizes, int n_in,
                              void* d_out, int out_size, void* d_ws, size_t ws_size,
                              hipStream_t stream) {
  (void)n_in; (void)out_size; (void)d_ws; (void)ws_size;
  const float* x1 = (const float*)d_in[0];
  const float* x2 = (const float*)d_in[1];
  const float* w  = (const float*)d_in[2];
  float* out = (float*)d_out;
  const int Z = in_sizes[0] / ZDIM_X1;
  const int grid = (Z + WAVES_PER_BLOCK - 1) / WAVES_PER_BLOCK;
  tp2_kernel<<<grid, WAVES_PER_BLOCK * 32, 0, stream>>>(x1, x2, w, out, Z);
}